// HydraInternal_39032662786563
// MI455X (gfx1250) — compile-verified
//
#include <hip/hip_runtime.h>

typedef __attribute__((ext_vector_type(2))) float v2f;
typedef __attribute__((ext_vector_type(8))) float v8f;

#define NUM_DIL 7
#define NBR     14          // NUM_DIL * 2 branches
#define LX      1024
#define FEATS   7168        // NBR * 512

// One workgroup per (sample, branch). 256 threads = 8 waves (wave32).
// Wave w computes channel tiles ct = 2w, 2w+1 (16 channels each) over all
// 64 time tiles. Full f32 path: 3 chained v_wmma_f32_16x16x4_f32 per tile
// (taps 0..8 in K-chunks 0-3 / 4-7 / 8-11, rest zero).
__global__ __launch_bounds__(256)
void hydra_kernel(const float* __restrict__ X,
                  const float* __restrict__ W,
                  float* __restrict__ out) {
  const int b    = blockIdx.x;
  const int n    = b / NBR;
  const int br   = b % NBR;        // br = di*2 + diff
  const int di   = br >> 1;
  const int diff = br & 1;
  const int d    = 1 << di;        // dilation
  const int Lin  = LX - diff;      // conv output length == input length
  const int pad  = 4 * d;          // symmetric padding (8*d)/2
  const int Lext = Lin + 8 * d;    // staged extent; lds idx = t + k*d

  __shared__ float sh[LX + 512 + 32];   // max extent 1536 + slack

  const int tid = threadIdx.x;
  const float* xrow = X + (size_t)n * LX;
  __builtin_prefetch(xrow, 0, 3);

  // Stage input (or first difference) with zero halos: sh[i] = input[i-pad].
  for (int i = tid; i < Lext + 16; i += 256) {
    const int src = i - pad;
    float v = 0.f;
    if (src >= 0 && src < Lin)
      v = diff ? (xrow[src + 1] - xrow[src]) : xrow[src];
    sh[i] = v;
  }
  __syncthreads();

  const int wave = tid >> 5;
  const int lane = tid & 31;
  const int l16  = lane & 15;      // matrix row (A) / column (B) within tile
  const int half = lane >> 4;
  const int koff = 2 * half;       // f32 16x4 layout: VGPR e <-> K = e + 2*half

  // A fragments: 2 channel tiles x 3 K-chunks, built once.
  const float* wbank = W + (size_t)br * 256 * 9;   // W[di][diff][c][0][tap]
  v2f afrag[2][3];
#pragma unroll
  for (int ci = 0; ci < 2; ++ci) {
    const int ct = wave * 2 + ci;
    const float* wrow = wbank + (size_t)(ct * 16 + l16) * 9;
#pragma unroll
    for (int c = 0; c < 3; ++c)
#pragma unroll
      for (int e = 0; e < 2; ++e) {
        const int k = 4 * c + e + koff;
        afrag[ci][c][e] = (k < 9) ? wrow[k] : 0.f;
      }
  }

  float cmax[2][8], cmin[2][8];
#pragma unroll
  for (int ci = 0; ci < 2; ++ci)
#pragma unroll
    for (int r = 0; r < 8; ++r) { cmax[ci][r] = 0.f; cmin[ci][r] = 0.f; }

  for (int tt = 0; tt < 64; ++tt) {
    const int  t      = tt * 16 + l16;   // time position for this lane's column
    const bool tvalid = t < Lin;

    // B fragments: B[k][col] = staged_input[t + k*d]; address clamped so the
    // load is always in-bounds, value masked for k >= 9.
    v2f bfrag[3];
#pragma unroll
    for (int c = 0; c < 3; ++c)
#pragma unroll
      for (int e = 0; e < 2; ++e) {
        const int k  = 4 * c + e + koff;
        const int kk = (k < 9) ? k : 8;
        const float v = sh[t + kk * d];
        bfrag[c][e] = (k < 9) ? v : 0.f;
      }

#pragma unroll
    for (int ci = 0; ci < 2; ++ci) {
      v8f acc = {};
#pragma unroll
      for (int c = 0; c < 3; ++c)
        acc = __builtin_amdgcn_wmma_f32_16x16x4_f32(
            /*neg_a=*/false, afrag[ci][c], /*neg_b=*/false, bfrag[c],
            /*c_mod=*/(short)0, acc, /*reuse_a=*/false, /*reuse_b=*/false);

      // This lane's 8 accumulator regs = 8 channels of group g = 2*ct + half
      // at time t. Tournament argmax/argmin, left-priority => first index.
      float vx[4], vn[4]; int ix[4], in_[4];
#pragma unroll
      for (int j = 0; j < 4; ++j) {
        const float a0 = acc[2 * j], a1 = acc[2 * j + 1];
        const bool cx = a1 > a0;  vx[j] = cx ? a1 : a0;  ix[j]  = cx ? 2 * j + 1 : 2 * j;
        const bool cn = a1 < a0;  vn[j] = cn ? a1 : a0;  in_[j] = cn ? 2 * j + 1 : 2 * j;
      }
      const bool cx0 = vx[1] > vx[0], cx1 = vx[3] > vx[2];
      const float mx0 = cx0 ? vx[1] : vx[0], mx1 = cx1 ? vx[3] : vx[2];
      const int   jx0 = cx0 ? ix[1] : ix[0], jx1 = cx1 ? ix[3] : ix[2];
      const bool cxf = mx1 > mx0;
      const float mv = cxf ? mx1 : mx0;
      const int   mi = cxf ? jx1 : jx0;

      const bool cn0 = vn[1] < vn[0], cn1 = vn[3] < vn[2];
      const float mn0 = cn0 ? vn[1] : vn[0], mn1 = cn1 ? vn[3] : vn[2];
      const int   jn0 = cn0 ? in_[1] : in_[0], jn1 = cn1 ? in_[3] : in_[2];
      const bool cnf = mn1 < mn0;
      const int   ni = cnf ? jn1 : jn0;

      const float addv = tvalid ? mv  : 0.f;   // junk column must be masked:
      const float addc = tvalid ? 1.f : 0.f;   // low taps read real data there
#pragma unroll
      for (int r = 0; r < 8; ++r) {            // branchless scatter-add
        cmax[ci][r] += (mi == r) ? addv : 0.f;
        cmin[ci][r] += (ni == r) ? addc : 0.f;
      }
    }
  }

  // Sum partials across the 16 lanes sharing each group (xor 1,2,4,8 stays
  // within each 16-lane half on wave32).
#pragma unroll
  for (int ci = 0; ci < 2; ++ci)
#pragma unroll
    for (int r = 0; r < 8; ++r) {
      float a = cmax[ci][r], m = cmin[ci][r];
#pragma unroll
      for (int mask = 1; mask < 16; mask <<= 1) {
        a += __shfl_xor(a, mask, 32);
        m += __shfl_xor(m, mask, 32);
      }
      cmax[ci][r] = a; cmin[ci][r] = m;
    }

  if (l16 == 0) {   // lanes 0 and 16 hold full sums for groups 2ct, 2ct+1
    float* obase = out + (size_t)n * FEATS + br * 512;
#pragma unroll
    for (int ci = 0; ci < 2; ++ci) {
      const int g = (wave * 2 + ci) * 2 + half;
#pragma unroll
      for (int r = 0; r < 8; ++r) {
        obase[g * 8 + r]       = cmax[ci][r];   // count_max block
        obase[256 + g * 8 + r] = cmin[ci][r];   // count_min block
      }
    }
  }
}

extern "C" void kernel_launch(void* const* d_in, const int* in_sizes, int n_in,
                              void* d_out, int out_size, void* d_ws, size_t ws_size,
                              hipStream_t stream) {
  const float* X = (const float*)d_in[0];   // (N, 1, 1024) f32
  const float* W = (const float*)d_in[1];   // (7, 2, 256, 1, 9) f32, pre-normalized
  float* out = (float*)d_out;               // (N, 7168) f32
  const int N = in_sizes[0] / LX;
  hydra_kernel<<<dim3(N * NBR), dim3(256), 0, stream>>>(X, W, out);
}